// GCNEncoder_18098992185810
// MI455X (gfx1250) — compile-verified
//
#include <hip/hip_runtime.h>
#include <hip/hip_bf16.h>

typedef float v2f __attribute__((ext_vector_type(2)));
typedef float v8f __attribute__((ext_vector_type(8)));

#define IN_C 256
#define HID  128
#define OUT_C 64

// ---------------- utility kernels ----------------
__global__ void k_fill(float* __restrict__ p, float v, long long n) {
  long long i = (long long)blockIdx.x * blockDim.x + threadIdx.x;
  if (i < n) p[i] = v;
}

__global__ void k_deg_count(const long long* __restrict__ dst, float* __restrict__ deg, int E) {
  int e = blockIdx.x * blockDim.x + threadIdx.x;
  if (e < E) atomicAdd(&deg[dst[e]], 1.0f);
}

__global__ void k_rsqrt(float* __restrict__ p, int n) {
  int i = blockIdx.x * blockDim.x + threadIdx.x;
  if (i < n) p[i] = rsqrtf(p[i]);
}

// ---------------- WMMA GEMM: C[M,NCOLS] = A[M,K] @ B[K,NCOLS] ----------------
// One wave per 16-row strip of A; all NCOLS/16 column tiles accumulated in
// registers so each A row is loaded exactly once. f32 WMMA (16x16x4) to match
// the fp32 reference precision.
template<int K, int NCOLS>
__global__ void k_gemm_wmma(const float* __restrict__ A, const float* __restrict__ B,
                            float* __restrict__ C, int M) {
  constexpr int NT = NCOLS / 16;
  const int wave = blockIdx.x * (blockDim.x >> 5) + (threadIdx.x >> 5);
  const int lane = threadIdx.x & 31;
  const int mtiles = M >> 4;           // M divisible by 16 (100000 / 16 = 6250)
  if (wave >= mtiles) return;          // wave-uniform: EXEC stays all-ones
  const int m0   = wave << 4;
  const int half = lane >> 4;          // 0: K=k,k+1 ; 1: K=k+2,k+3
  const int l16  = lane & 15;

  const float* arow = A + (size_t)(m0 + l16) * K;

  v8f acc[NT];
#pragma unroll
  for (int t = 0; t < NT; ++t) acc[t] = (v8f){0.f,0.f,0.f,0.f,0.f,0.f,0.f,0.f};

  for (int k = 0; k < K; k += 4) {
    const int ka = k + half * 2;
    v2f a;
    a.x = arow[ka];
    a.y = arow[ka + 1];
#pragma unroll
    for (int t = 0; t < NT; ++t) {
      const float* bp = B + (size_t)ka * NCOLS + t * 16 + l16;
      v2f b;
      b.x = bp[0];
      b.y = bp[NCOLS];
      acc[t] = __builtin_amdgcn_wmma_f32_16x16x4_f32(
          false, a, false, b, (short)0, acc[t], false, false);
    }
  }

  // D layout: VGPR r -> row m0 + r + half*8, col = t*16 + l16
#pragma unroll
  for (int t = 0; t < NT; ++t) {
    float* cp = C + (size_t)(m0 + half * 8) * NCOLS + t * 16 + l16;
#pragma unroll
    for (int r = 0; r < 8; ++r) cp[(size_t)r * NCOLS] = acc[t][r];
  }
}

// ---------------- edge scatter: Hacc[dst] += norm * Hp[src] ----------------
// One wave per edge; each lane owns CH/32 contiguous channels (vectorized
// gather, per-float f32 atomics into L2-resident accumulator).
template<int CH>
__global__ void k_scatter(const long long* __restrict__ src, const long long* __restrict__ dst,
                          const float* __restrict__ dis, const float* __restrict__ Hp,
                          float* __restrict__ Hacc, int E) {
  constexpr int PL = CH / 32;
  long long tid = (long long)blockIdx.x * blockDim.x + threadIdx.x;
  int e = (int)(tid >> 5);
  if (e >= E) return;
  int lane = (int)(tid & 31);
  long long s = src[e], d = dst[e];
  float norm = dis[s] * dis[d];
  const float* hp  = Hp   + (size_t)s * CH + lane * PL;
  float*       out = Hacc + (size_t)d * CH + lane * PL;
  float v[PL];
#pragma unroll
  for (int j = 0; j < PL; ++j) v[j] = hp[j];
#pragma unroll
  for (int j = 0; j < PL; ++j) atomicAdd(out + j, v[j] * norm);
}

// ---------------- epilogue: out = [relu](Hacc + dis^2 * Hp + bias) ----------------
template<int CH, bool RELU>
__global__ void k_finish(const float* __restrict__ Hacc, const float* __restrict__ Hp,
                         const float* __restrict__ dis, const float* __restrict__ bias,
                         float* __restrict__ out, int Mn) {
  long long i = (long long)blockIdx.x * blockDim.x + threadIdx.x;
  if (i >= (long long)Mn * CH) return;
  int node = (int)(i / CH);
  int c    = (int)(i % CH);
  float di = dis[node];
  float v  = Hacc[i] + di * di * Hp[i] + bias[c];
  out[i] = RELU ? fmaxf(v, 0.0f) : v;
}

extern "C" void kernel_launch(void* const* d_in, const int* in_sizes, int n_in,
                              void* d_out, int out_size, void* d_ws, size_t ws_size,
                              hipStream_t stream) {
  const float*     x   = (const float*)d_in[0];
  const long long* ei  = (const long long*)d_in[1];   // int64 [2, E]
  const float*     W1  = (const float*)d_in[2];
  const float*     b1  = (const float*)d_in[3];
  const float*     W2  = (const float*)d_in[4];
  const float*     b2  = (const float*)d_in[5];
  float*           out = (float*)d_out;

  const int Nn = in_sizes[0] / IN_C;   // 100000
  const int E  = in_sizes[1] / 2;      // 1600000
  const long long* srcIdx = ei;
  const long long* dstIdx = ei + E;

  // workspace layout (all fits comfortably in the 192MB L2):
  //   dis  : Nn floats
  //   H1p  : Nn*HID floats   (layer-1 pre-aggregation features)
  //   H1   : Nn*HID floats   (layer-1 aggregated / activated)
  //   H2p  : aliases H1p[0 .. Nn*OUT_C)       (H1p dead by then)
  //   H2a  : aliases H1p[Nn*OUT_C .. 2*Nn*OUT_C)
  float* dis = (float*)d_ws;
  size_t disPad = ((size_t)Nn + 63) & ~(size_t)63;
  float* H1p = dis + disPad;
  float* H1  = H1p + (size_t)Nn * HID;
  float* H2p = H1p;
  float* H2a = H1p + (size_t)Nn * OUT_C;

  const int T = 256;
  const int WPB = T / 32;  // waves per block

  // ---- degree / normalization: deg = 1 (self-loop) + in-degree; dis = rsqrt(deg)
  k_fill<<<(Nn + T - 1) / T, T, 0, stream>>>(dis, 1.0f, (long long)Nn);
  k_deg_count<<<(E + T - 1) / T, T, 0, stream>>>(dstIdx, dis, E);
  k_rsqrt<<<(Nn + T - 1) / T, T, 0, stream>>>(dis, Nn);

  // ---- layer 1: H1p = X @ W1 ; H1 = relu(scatter(H1p) + dis^2*H1p + b1)
  {
    int waves = Nn >> 4;
    k_gemm_wmma<IN_C, HID><<<(waves + WPB - 1) / WPB, T, 0, stream>>>(x, W1, H1p, Nn);
  }
  {
    long long n = (long long)Nn * HID;
    k_fill<<<(int)((n + T - 1) / T), T, 0, stream>>>(H1, 0.0f, n);
  }
  {
    long long threads = (long long)E * 32;
    k_scatter<HID><<<(int)((threads + T - 1) / T), T, 0, stream>>>(srcIdx, dstIdx, dis, H1p, H1, E);
  }
  {
    long long n = (long long)Nn * HID;
    k_finish<HID, true><<<(int)((n + T - 1) / T), T, 0, stream>>>(H1, H1p, dis, b1, H1, Nn);
  }

  // ---- layer 2: H2p = H1 @ W2 ; out = scatter(H2p) + dis^2*H2p + b2
  {
    int waves = Nn >> 4;
    k_gemm_wmma<HID, OUT_C><<<(waves + WPB - 1) / WPB, T, 0, stream>>>(H1, W2, H2p, Nn);
  }
  {
    long long n = (long long)Nn * OUT_C;
    k_fill<<<(int)((n + T - 1) / T), T, 0, stream>>>(H2a, 0.0f, n);
  }
  {
    long long threads = (long long)E * 32;
    k_scatter<OUT_C><<<(int)((threads + T - 1) / T), T, 0, stream>>>(srcIdx, dstIdx, dis, H2p, H2a, E);
  }
  {
    long long n = (long long)Nn * OUT_C;
    k_finish<OUT_C, false><<<(int)((n + T - 1) / T), T, 0, stream>>>(H2a, H2p, dis, b2, out, Nn);
  }
}